// TemporalHeteroConv_76312978915561
// MI455X (gfx1250) — compile-verified
//
#include <hip/hip_runtime.h>
#include <hip/hip_bf16.h>
#include <math.h>

// ---------------------------------------------------------------------------
// TemporalHeteroConv (TGAT-style) for MI455X / gfx1250, wave32.
// Dense projections: V_WMMA_F32_16X16X4_F32.
// Edge aggregation: CSR build (histogram + scan + fill), then wave-per-node
// gather with in-register softmax + accumulation (no float atomics).
// ---------------------------------------------------------------------------

#define N_NODES 100000
#define N_EDGES 800000
#define CCH 64            // channels
#define NH  4             // heads
#define CH  (CCH * NH)    // 256
#define SCAN_B 391        // ceil(N_NODES / 256)

typedef __attribute__((ext_vector_type(2))) float v2f;
typedef __attribute__((ext_vector_type(8))) float v8f;

// ---------------------------------------------------------------------------
__global__ void k_zero(int* __restrict__ p, int n) {
  int i = blockIdx.x * blockDim.x + threadIdx.x;
  if (i < n) p[i] = 0;
}

// ---------------------------------------------------------------------------
// h = x @ W_w + W_b          (N x 64) = (N x 64)(64 x 64)
// one wave -> one 16x16 tile; K swept in steps of 4 via f32 WMMA.
// A 16x4 frag: lanes 0-15 hold K=0,1; lanes 16-31 hold K=2,3 (ISA 7.12.2).
// ---------------------------------------------------------------------------
__global__ void k_gemm_h(const float* __restrict__ x, const float* __restrict__ W,
                         const float* __restrict__ bias, float* __restrict__ h) {
  const int lane    = threadIdx.x & 31;
  const int wave    = threadIdx.x >> 5;
  const int colBase = (wave & 3) * 16;
  const int rowBase = (blockIdx.x * 2 + (wave >> 2)) * 16;
  const int lh = lane & 15;
  const int kk = (lane >> 4) << 1;  // 0 or 2

  v8f acc = {};
  const float* arow = x + (size_t)(rowBase + lh) * CCH;
#pragma unroll
  for (int k = 0; k < CCH; k += 4) {
    v2f a, b;
    a.x = arow[k + kk];
    a.y = arow[k + kk + 1];
    b.x = W[(k + kk)     * CCH + colBase + lh];
    b.y = W[(k + kk + 1) * CCH + colBase + lh];
    acc = __builtin_amdgcn_wmma_f32_16x16x4_f32(false, a, false, b,
                                                (short)0, acc, false, false);
  }
  const float bv = bias[colBase + lh];
  const int mBase = (lane < 16) ? 0 : 8;
#pragma unroll
  for (int i = 0; i < 8; ++i)
    h[(size_t)(rowBase + mBase + i) * CCH + colBase + lh] = acc[i] + bv;
}

// ---------------------------------------------------------------------------
// CSR build: degree histogram -> 2-level exclusive scan -> fill
// ---------------------------------------------------------------------------
__global__ void k_degree(const int* __restrict__ ei, int* __restrict__ deg) {
  int e = blockIdx.x * blockDim.x + threadIdx.x;
  if (e < N_EDGES) atomicAdd(&deg[ei[N_EDGES + e]], 1);
}

__global__ void k_scan1(const int* __restrict__ deg, int* __restrict__ bsum) {
  __shared__ int tmp[256];
  const int t = threadIdx.x;
  const int i = blockIdx.x * 256 + t;
  tmp[t] = (i < N_NODES) ? deg[i] : 0;
  __syncthreads();
#pragma unroll
  for (int off = 128; off >= 1; off >>= 1) {
    if (t < off) tmp[t] += tmp[t + off];
    __syncthreads();
  }
  if (t == 0) bsum[blockIdx.x] = tmp[0];
}

__global__ void k_scan2(int* __restrict__ bsum) {  // tiny serial exclusive scan
  if (blockIdx.x == 0 && threadIdx.x == 0) {
    int acc = 0;
    for (int b = 0; b < SCAN_B; ++b) { int v = bsum[b]; bsum[b] = acc; acc += v; }
  }
}

__global__ void k_scan3(const int* __restrict__ deg, const int* __restrict__ bsum,
                        int* __restrict__ rowptr, int* __restrict__ cursor) {
  __shared__ int tmp[256];
  const int t = threadIdx.x;
  const int i = blockIdx.x * 256 + t;
  const int v = (i < N_NODES) ? deg[i] : 0;
  tmp[t] = v;
  __syncthreads();
#pragma unroll
  for (int off = 1; off < 256; off <<= 1) {   // Hillis-Steele inclusive scan
    int a = (t >= off) ? tmp[t - off] : 0;
    __syncthreads();
    tmp[t] += a;
    __syncthreads();
  }
  const int ex = tmp[t] - v + bsum[blockIdx.x];
  if (i < N_NODES) { rowptr[i] = ex; cursor[i] = ex; }
  if (i == 0) rowptr[N_NODES] = N_EDGES;
}

__global__ void k_fill(const int* __restrict__ ei, int* __restrict__ cursor,
                       int* __restrict__ esort) {
  int e = blockIdx.x * blockDim.x + threadIdx.x;
  if (e < N_EDGES) {
    int pos = atomicAdd(&cursor[ei[N_EDGES + e]], 1);
    esort[pos] = e;
  }
}

// ---------------------------------------------------------------------------
// Per-edge attention logits -> leakyrelu -> temporal decay -> scores.
// One wave per edge; each lane owns channels {lane, lane+32}; 4-head partials
// reduced with wave32 xor-shuffles. No atomics.
// ---------------------------------------------------------------------------
__global__ void k_edge_scores(const float* __restrict__ h, const int* __restrict__ ei,
                              const float* __restrict__ et, const int* __restrict__ ct,
                              const float* __restrict__ aw, const float* __restrict__ ab,
                              const float* __restrict__ dr, float* __restrict__ scores) {
  const int e = (int)((blockIdx.x * blockDim.x + threadIdx.x) >> 5);
  if (e >= N_EDGES) return;
  const int lane = threadIdx.x & 31;
  const int src = ei[e];
  const int dst = ei[N_EDGES + e];
  const int c0 = lane, c1 = lane + 32;

  const float hs0 = h[(size_t)src * CCH + c0], hs1 = h[(size_t)src * CCH + c1];
  const float hd0 = h[(size_t)dst * CCH + c0], hd1 = h[(size_t)dst * CCH + c1];

  float acc[NH];
#pragma unroll
  for (int j = 0; j < NH; ++j) {
    acc[j] = hs0 * aw[c0 * NH + j] + hs1 * aw[c1 * NH + j] +
             hd0 * aw[(CCH + c0) * NH + j] + hd1 * aw[(CCH + c1) * NH + j];
  }
#pragma unroll
  for (int j = 0; j < NH; ++j)
#pragma unroll
    for (int off = 16; off >= 1; off >>= 1)
      acc[j] += __shfl_xor(acc[j], off, 32);

  if (lane == 0) {
    const float lam   = log1pf(expf(dr[0]));                 // softplus
    const float decay = expf(-lam * ((float)ct[0] - et[e]));
#pragma unroll
    for (int j = 0; j < NH; ++j) {
      float l = acc[j] + ab[j];
      scores[(size_t)e * NH + j] = (l > 0.0f ? l : 0.2f * l) * decay;
    }
  }
}

// ---------------------------------------------------------------------------
// Wave-per-destination gather: segmented softmax + weighted accumulation,
// all 256 agg accumulators held in registers (8 per lane), plain stores out.
// ---------------------------------------------------------------------------
__global__ void k_aggregate(const float* __restrict__ h, const int* __restrict__ ei,
                            const float* __restrict__ scores,
                            const int* __restrict__ rowptr, const int* __restrict__ esort,
                            float* __restrict__ agg) {
  const int n = (int)((blockIdx.x * blockDim.x + threadIdx.x) >> 5);
  if (n >= N_NODES) return;
  const int lane = threadIdx.x & 31;
  const int beg = rowptr[n], end = rowptr[n + 1];

  // phase 1: per-head segment max (lane-strided over edges)
  float mx[NH] = {-3.402823466e+38f, -3.402823466e+38f,
                  -3.402823466e+38f, -3.402823466e+38f};
  for (int i = beg + lane; i < end; i += 32) {
    const int e = esort[i];
#pragma unroll
    for (int j = 0; j < NH; ++j) mx[j] = fmaxf(mx[j], scores[(size_t)e * NH + j]);
  }
#pragma unroll
  for (int j = 0; j < NH; ++j)
#pragma unroll
    for (int off = 16; off >= 1; off >>= 1)
      mx[j] = fmaxf(mx[j], __shfl_xor(mx[j], off, 32));

  // phase 2: per-head sum of exp(s - max)
  float sm[NH] = {0.0f, 0.0f, 0.0f, 0.0f};
  for (int i = beg + lane; i < end; i += 32) {
    const int e = esort[i];
#pragma unroll
    for (int j = 0; j < NH; ++j) sm[j] += expf(scores[(size_t)e * NH + j] - mx[j]);
  }
#pragma unroll
  for (int j = 0; j < NH; ++j)
#pragma unroll
    for (int off = 16; off >= 1; off >>= 1)
      sm[j] += __shfl_xor(sm[j], off, 32);

  float inv[NH];
#pragma unroll
  for (int j = 0; j < NH; ++j) inv[j] = 1.0f / (sm[j] + 1e-8f);

  // phase 3: accumulate messages; whole wave cooperates per edge
  float acc[2 * NH] = {0};
  for (int i = beg; i < end; ++i) {
    const int e = esort[i];
    const int src = ei[e];
    const float hs0 = h[(size_t)src * CCH + lane];
    const float hs1 = h[(size_t)src * CCH + lane + 32];
#pragma unroll
    for (int j = 0; j < NH; ++j) {
      const float a = expf(scores[(size_t)e * NH + j] - mx[j]) * inv[j];
      acc[j]      += hs0 * a;
      acc[NH + j] += hs1 * a;
    }
  }
  float* arow = agg + (size_t)n * CH;
#pragma unroll
  for (int j = 0; j < NH; ++j) {
    arow[lane * NH + j]        = acc[j];
    arow[(lane + 32) * NH + j] = acc[NH + j];
  }
}

// ---------------------------------------------------------------------------
// z = agg @ proj_w + proj_b + x      (N x 64) = (N x 256)(256 x 64)
// ---------------------------------------------------------------------------
__global__ void k_gemm_out(const float* __restrict__ agg, const float* __restrict__ Pw,
                           const float* __restrict__ pb, const float* __restrict__ x,
                           float* __restrict__ z) {
  const int lane    = threadIdx.x & 31;
  const int wave    = threadIdx.x >> 5;
  const int colBase = (wave & 3) * 16;
  const int rowBase = (blockIdx.x * 2 + (wave >> 2)) * 16;
  const int lh = lane & 15;
  const int kk = (lane >> 4) << 1;

  v8f acc = {};
  const float* arow = agg + (size_t)(rowBase + lh) * CH;
#pragma unroll 4
  for (int k = 0; k < CH; k += 4) {
    v2f a, b;
    a.x = arow[k + kk];
    a.y = arow[k + kk + 1];
    b.x = Pw[(k + kk)     * CCH + colBase + lh];
    b.y = Pw[(k + kk + 1) * CCH + colBase + lh];
    acc = __builtin_amdgcn_wmma_f32_16x16x4_f32(false, a, false, b,
                                                (short)0, acc, false, false);
  }
  const float bv = pb[colBase + lh];
  const int mBase = (lane < 16) ? 0 : 8;
#pragma unroll
  for (int i = 0; i < 8; ++i) {
    size_t idx = (size_t)(rowBase + mBase + i) * CCH + colBase + lh;
    z[idx] = acc[i] + bv + x[idx];
  }
}

// ---------------------------------------------------------------------------
// y = relu(LayerNorm(z)). One wave per row (64 ch -> 2 per lane).
// ---------------------------------------------------------------------------
__global__ void k_ln_relu(const float* __restrict__ z, const float* __restrict__ g,
                          const float* __restrict__ b, float* __restrict__ out) {
  const int r = (int)((blockIdx.x * blockDim.x + threadIdx.x) >> 5);
  if (r >= N_NODES) return;
  const int lane = threadIdx.x & 31;
  float z0 = z[(size_t)r * CCH + lane];
  float z1 = z[(size_t)r * CCH + lane + 32];

  float s = z0 + z1;
#pragma unroll
  for (int off = 16; off >= 1; off >>= 1) s += __shfl_xor(s, off, 32);
  float mu = s * (1.0f / 64.0f);
  float d0 = z0 - mu, d1 = z1 - mu;
  float v = d0 * d0 + d1 * d1;
#pragma unroll
  for (int off = 16; off >= 1; off >>= 1) v += __shfl_xor(v, off, 32);
  float inv = rsqrtf(v * (1.0f / 64.0f) + 1e-5f);

  float y0 = g[lane] * d0 * inv + b[lane];
  float y1 = g[lane + 32] * d1 * inv + b[lane + 32];
  out[(size_t)r * CCH + lane]      = y0 > 0.0f ? y0 : 0.0f;
  out[(size_t)r * CCH + lane + 32] = y1 > 0.0f ? y1 : 0.0f;
}

// ---------------------------------------------------------------------------
extern "C" void kernel_launch(void* const* d_in, const int* in_sizes, int n_in,
                              void* d_out, int out_size, void* d_ws, size_t ws_size,
                              hipStream_t stream) {
  const float* x  = (const float*)d_in[0];
  const int*   ei = (const int*)  d_in[1];
  const float* et = (const float*)d_in[2];
  const int*   ct = (const int*)  d_in[3];
  const float* Ww = (const float*)d_in[4];
  const float* Wb = (const float*)d_in[5];
  const float* aw = (const float*)d_in[6];
  const float* ab = (const float*)d_in[7];
  const float* dr = (const float*)d_in[8];
  const float* pw = (const float*)d_in[9];
  const float* pb = (const float*)d_in[10];
  const float* lg = (const float*)d_in[11];
  const float* lb = (const float*)d_in[12];
  float* out = (float*)d_out;

  // workspace layout (bytes):
  //   h      [N*64 f32]    @ 0            (25,600,000)
  //   scores [E*4  f32]    @ 25,600,000   (12,800,000)
  //   deg    [N    i32]    @ 38,400,000   (   400,000)
  //   rowptr [N+1  i32]    @ 38,800,000   (   400,512 padded)
  //   cursor [N    i32]    @ 39,200,512   (   400,000)
  //   esort  [E    i32]    @ 39,600,512   ( 3,200,000)
  //   bsum   [391  i32]    @ 42,800,512   (     2,048 padded)
  //   agg    [N*256 f32]   @ 42,802,560   (102,400,000)
  //   z      [N*64 f32]    @ 145,202,560  (25,600,000)   total ~170.8 MB
  char* ws = (char*)d_ws;
  float* h      = (float*)(ws + 0);
  float* scores = (float*)(ws + 25600000);
  int*   deg    = (int*)  (ws + 38400000);
  int*   rowptr = (int*)  (ws + 38800000);
  int*   cursor = (int*)  (ws + 39200512);
  int*   esort  = (int*)  (ws + 39600512);
  int*   bsum   = (int*)  (ws + 42800512);
  float* agg    = (float*)(ws + 42802560);
  float* z      = (float*)(ws + 145202560);

  // 1) zero degree histogram only (everything else fully overwritten)
  k_zero<<<SCAN_B, 256, 0, stream>>>(deg, N_NODES);
  // 2) h = x @ W + b  (WMMA)
  k_gemm_h<<<N_NODES / 32, 256, 0, stream>>>(x, Ww, Wb, h);
  // 3) CSR build
  k_degree<<<N_EDGES / 256, 256, 0, stream>>>(ei, deg);
  k_scan1<<<SCAN_B, 256, 0, stream>>>(deg, bsum);
  k_scan2<<<1, 64, 0, stream>>>(bsum);
  k_scan3<<<SCAN_B, 256, 0, stream>>>(deg, bsum, rowptr, cursor);
  k_fill<<<N_EDGES / 256, 256, 0, stream>>>(ei, cursor, esort);
  // 4) per-edge attention scores (1 edge per wave)
  k_edge_scores<<<N_EDGES / 8, 256, 0, stream>>>(h, ei, et, ct, aw, ab, dr, scores);
  // 5) per-node softmax + gather-accumulate (1 node per wave, no float atomics)
  k_aggregate<<<(N_NODES + 7) / 8, 256, 0, stream>>>(h, ei, scores, rowptr, esort, agg);
  // 6) z = agg @ proj_w + proj_b + x  (WMMA)
  k_gemm_out<<<N_NODES / 32, 256, 0, stream>>>(agg, pw, pb, x, z);
  // 7) LayerNorm + ReLU -> out
  k_ln_relu<<<N_NODES / 8, 256, 0, stream>>>(z, lg, lb, out);
}